// MultiBodyConvolution_45707041964607
// MI455X (gfx1250) — compile-verified
//
#include <hip/hip_runtime.h>

// Problem sizes (fixed by the reference)
#define NN   64
#define AA   256
#define BB   128
#define NRBF 32
#define HID  32

typedef __attribute__((ext_vector_type(2))) float v2f;
typedef __attribute__((ext_vector_type(8))) float v8f;

__device__ __forceinline__ float silu_f(float x) {
    // x * sigmoid(x) using native v_exp_f32 + v_rcp_f32 (avoids the IEEE
    // v_div_scale/v_div_fmas/v_div_fixup chain the plain '/' emits).
    return x * __builtin_amdgcn_rcpf(1.0f + __expf(-x));
}

// One block (8 wave32) per (n,a). Each wave handles a 16-neighbor tile.
// GEMM: X[128,32] @ W1[32,32] via V_WMMA_F32_16X16X4_F32 (8 K-steps, 2 N-tiles).
__global__ __launch_bounds__(256) void MultiBodyConvolution_45707041964607_kernel(
    const float* __restrict__ r,      // [N,A,B,3]
    const float* __restrict__ rbfs,   // [N,A,B,NRBF]
    const float* __restrict__ f_old,  // [N,A,3]
    const float* __restrict__ W1,     // [NRBF,HID]
    const float* __restrict__ b1,     // [HID]
    const float* __restrict__ W2,     // [HID] (HID x 1)
    const float* __restrict__ b2,     // [1]
    float* __restrict__ out)          // [N,A,3]
{
    const int na   = blockIdx.x;       // 0 .. N*A-1
    const int tid  = threadIdx.x;      // 0 .. 255
    const int wave = tid >> 5;         // 0 .. 7
    const int lane = tid & 31;
    const int half = lane >> 4;        // 0 or 1 (selects K pair / row group)
    const int l15  = lane & 15;        // row (A frag) / column (B,C,D frags)

    __shared__ float s_lds[BB];        // gate value per neighbor
    __shared__ float acc[3];           // output accumulator

    if (tid < 3) acc[tid] = 0.0f;

    const int rowbase = wave * 16;     // first neighbor of this wave's tile

    // ---- A fragments: 16x4 f32 tiles of rbfs -------------------------------
    // ISA layout (32-bit A, 16x4): lanes 0-15 = rows, v0 holds K=0 (half 0) /
    // K=2 (half 1), v1 holds K=1 / K=3.  8 K-chunks cover NRBF=32.
    const float* xrow = rbfs + ((size_t)na * BB + (rowbase + l15)) * NRBF + 2 * half;
    v2f afrag[8];
#pragma unroll
    for (int kc = 0; kc < 8; ++kc) {
        afrag[kc].x = xrow[4 * kc + 0];
        afrag[kc].y = xrow[4 * kc + 1];
    }

    // ---- B fragments: 4x16 f32 tiles of W1 (column-major per lane) ---------
    v2f bfrag[2][8];
#pragma unroll
    for (int t = 0; t < 2; ++t) {
        const int col = l15 + 16 * t;
#pragma unroll
        for (int kc = 0; kc < 8; ++kc) {
            const int k = 4 * kc + 2 * half;
            bfrag[t][kc].x = W1[(k + 0) * HID + col];
            bfrag[t][kc].y = W1[(k + 1) * HID + col];
        }
    }

    // ---- GEMM: 16 x v_wmma_f32_16x16x4_f32 per wave ------------------------
    v8f h0 = {};
    v8f h1 = {};
#pragma unroll
    for (int kc = 0; kc < 8; ++kc) {
        h0 = __builtin_amdgcn_wmma_f32_16x16x4_f32(
            false, afrag[kc], false, bfrag[0][kc], (short)0, h0, false, false);
        h1 = __builtin_amdgcn_wmma_f32_16x16x4_f32(
            false, afrag[kc], false, bfrag[1][kc], (short)0, h1, false, false);
    }

    // ---- Epilogue: bias + silu + dot with W2 -------------------------------
    // C/D layout: lane (l15) = column N, element i = row (i + 8*half).
    const float bias0 = b1[l15];
    const float bias1 = b1[l15 + 16];
    const float w20   = W2[l15];
    const float w21   = W2[l15 + 16];
    v8f part;
#pragma unroll
    for (int i = 0; i < 8; ++i) {
        const float a0 = h0[i] + bias0;
        const float a1 = h1[i] + bias1;
        part[i] = silu_f(a0) * w20 + silu_f(a1) * w21;
    }
    // Reduce across the 16 lanes of each half (sums over all 32 hidden cols).
#pragma unroll
    for (int m = 1; m < 16; m <<= 1) {
#pragma unroll
        for (int i = 0; i < 8; ++i)
            part[i] += __shfl_xor(part[i], m, 32);
    }

    const float b2v = b2[0];
    if (l15 == 0) {
#pragma unroll
        for (int i = 0; i < 8; ++i)
            s_lds[rowbase + 8 * half + i] = part[i] + b2v;
    }
    __syncthreads();

    // ---- Gated cross product + neighbor reduction --------------------------
    const float fx = f_old[3 * na + 0];
    const float fy = f_old[3 * na + 1];
    const float fz = f_old[3 * na + 2];

    float cx = 0.0f, cy = 0.0f, cz = 0.0f;
    if (tid < BB) {
        const float* rp = r + ((size_t)na * BB + tid) * 3;
        const float rx = rp[0], ry = rp[1], rz = rp[2];
        const float s  = s_lds[tid];
        cx = s * (ry * fz - rz * fy);
        cy = s * (rz * fx - rx * fz);
        cz = s * (rx * fy - ry * fx);
    }
#pragma unroll
    for (int m = 1; m < 32; m <<= 1) {
        cx += __shfl_xor(cx, m, 32);
        cy += __shfl_xor(cy, m, 32);
        cz += __shfl_xor(cz, m, 32);
    }
    if (lane == 0 && wave < 4) {   // only waves 0-3 held real neighbors
        atomicAdd(&acc[0], cx);
        atomicAdd(&acc[1], cy);
        atomicAdd(&acc[2], cz);
    }
    __syncthreads();

    if (tid < 3) out[3 * na + tid] = acc[tid];
}

extern "C" void kernel_launch(void* const* d_in, const int* in_sizes, int n_in,
                              void* d_out, int out_size, void* d_ws, size_t ws_size,
                              hipStream_t stream) {
    (void)in_sizes; (void)n_in; (void)out_size; (void)d_ws; (void)ws_size;
    // setup_inputs order: t(0, unused), r(1), rbfs(2), f_old(3), W1(4), b1(5), W2(6), b2(7)
    const float* r     = (const float*)d_in[1];
    const float* rbfs  = (const float*)d_in[2];
    const float* f_old = (const float*)d_in[3];
    const float* W1    = (const float*)d_in[4];
    const float* b1    = (const float*)d_in[5];
    const float* W2    = (const float*)d_in[6];
    const float* b2    = (const float*)d_in[7];
    float* out = (float*)d_out;

    dim3 grid(NN * AA);   // 16384 blocks, one per (n,a)
    dim3 block(256);      // 8 wave32
    MultiBodyConvolution_45707041964607_kernel<<<grid, block, 0, stream>>>(
        r, rbfs, f_old, W1, b1, W2, b2, out);
}